// GraphQNetwork_80479097192800
// MI455X (gfx1250) — compile-verified
//
#include <hip/hip_runtime.h>
#include <math.h>

#define NNODES 100000
#define NEDGES 1600000
#define NGRAPH 64
#define NSUB   50000

typedef float v8f __attribute__((ext_vector_type(8)));
typedef float v2f __attribute__((ext_vector_type(2)));

// ---------------------------------------------------------------- utilities
__global__ void k_fill(float* __restrict__ p, float v, int n) {
    int i = blockIdx.x * blockDim.x + threadIdx.x;
    if (i < n) p[i] = v;
}

// ---------------------------------------------------------------- graph norm
__global__ void k_gn_sum(const float* __restrict__ x, const int* __restrict__ batch,
                         float* gsum, float* gcnt, int n, int F) {
    int i = blockIdx.x * blockDim.x + threadIdx.x;
    if (i >= n) return;
    int b = batch[i];
    atomicAdd(&gcnt[b], 1.0f);
    for (int f = 0; f < F; ++f) atomicAdd(&gsum[b * F + f], x[i * F + f]);
}

__global__ void k_gn_mean(float* gsum, const float* gcnt, int G, int F) {
    int i = blockIdx.x * blockDim.x + threadIdx.x;
    if (i >= G * F) return;
    int g = i / F;
    gsum[i] /= fmaxf(gcnt[g], 1.0f);
}

// reads x (stride F), writes centered xc (padded stride SP)
__global__ void k_gn_var(const float* __restrict__ x, const int* __restrict__ batch,
                         const float* __restrict__ mean, const float* __restrict__ ms,
                         float* xc, float* gvar, int n, int F, int SP) {
    int i = blockIdx.x * blockDim.x + threadIdx.x;
    if (i >= n) return;
    int b = batch[i];
    for (int f = 0; f < F; ++f) {
        float v = x[i * F + f] - ms[f] * mean[b * F + f];
        xc[(size_t)i * SP + f] = v;
        atomicAdd(&gvar[b * F + f], v * v);
    }
}

__global__ void k_gn_fin(float* xc, const int* __restrict__ batch,
                         const float* __restrict__ gvar, const float* __restrict__ gcnt,
                         const float* __restrict__ w, const float* __restrict__ bb,
                         int n, int F, int SP) {
    int i = blockIdx.x * blockDim.x + threadIdx.x;
    if (i >= n) return;
    int b = batch[i];
    float cnt = fmaxf(gcnt[b], 1.0f);
    for (int f = 0; f < F; ++f) {
        float var = gvar[b * F + f] / cnt;
        xc[(size_t)i * SP + f] = w[f] * xc[(size_t)i * SP + f] * rsqrtf(var + 1e-5f) + bb[f];
    }
}

// ---------------------------------------------------------------- WMMA GEMM
// Y[M,Ncol] = X[M,Kpad(zero-padded)] @ W[K,Ncol] + bias ; M multiple of 16.
// Block = 256 threads (8 waves); blockIdx.y = 16-col tile; W tile staged in LDS
// pair-interleaved so B operand is one ds_load_b64. WMMA loop is branch-free.
__global__ void k_gemm_wmma(const float* __restrict__ X, const float* __restrict__ W,
                            const float* __restrict__ bias, float* __restrict__ Y,
                            int M, int K, int Kpad, int Ncol) {
    __shared__ float sW2[1920];                // up to Kpad=120 x 16 cols
    int col0 = (int)blockIdx.y << 4;
    // cooperative, zero-padded W tile load: sW2[(k>>1)*32 + c*2 + (k&1)] = W[k][col0+c]
    for (int idx = threadIdx.x; idx < Kpad * 16; idx += blockDim.x) {
        int k = idx >> 4, c = idx & 15;
        int col = col0 + c;
        float v = (k < K && col < Ncol) ? W[(size_t)k * Ncol + col] : 0.0f;
        sW2[((k >> 1) << 5) + (c << 1) + (k & 1)] = v;
    }
    __syncthreads();

    int wave = threadIdx.x >> 5;
    int lane = threadIdx.x & 31;
    int rowTile = blockIdx.x * (blockDim.x >> 5) + wave;
    if (rowTile >= (M >> 4)) return;           // wave-uniform
    int row0 = rowTile << 4;
    int m  = lane & 15;                        // A: row ; B/D: col
    int kh = lane >> 4;                        // half-wave selector

    const float* xrow = X + (size_t)(row0 + m) * Kpad;
    v8f acc = {0.f, 0.f, 0.f, 0.f, 0.f, 0.f, 0.f, 0.f};
    for (int kb = 0; kb < Kpad; kb += 4) {
        int k0 = kb + 2 * kh;
        v2f a  = *(const v2f*)(xrow + k0);                       // global_load_b64
        v2f bm = *(const v2f*)(&sW2[((k0 >> 1) << 5) + (m << 1)]); // ds_load_b64
        acc = __builtin_amdgcn_wmma_f32_16x16x4_f32(false, a, false, bm,
                                                    (short)0, acc, false, false);
    }
    int col = col0 + m;
    if (col < Ncol) {
        float bv = bias[col];
        for (int r = 0; r < 8; ++r)
            Y[(size_t)(row0 + r + 8 * kh) * Ncol + col] = acc[r] + bv;
    }
}

// ---------------------------------------------------------------- GATv2 edges
__device__ __forceinline__ void atomicMaxF(float* addr, float val) {
    if (val >= 0.0f) atomicMax((int*)addr, __float_as_int(val));
    else             atomicMin((unsigned int*)addr, (unsigned int)__float_as_int(val));
}

__global__ void k_edge_score(const float* __restrict__ xl, const float* __restrict__ xr,
                             const float* __restrict__ ea, const int* __restrict__ ei,
                             const float* __restrict__ We, const float* __restrict__ att,
                             float* esc, float* emax, int E, int H, int O) {
    int i = blockIdx.x * blockDim.x + threadIdx.x;
    if (i >= E * H) return;
    int e = i / H, h = i - e * H;
    int s = ei[e], d = ei[E + e];
    float eav = ea[e];
    int HO = H * O;
    const float* pl = xl + (size_t)s * HO + h * O;
    const float* pr = xr + (size_t)d * HO + h * O;
    const float* pw = We + h * O;
    const float* pa = att + h * O;
    float acc = 0.0f;
    for (int o = 0; o < O; ++o) {
        float v = pl[o] + pr[o] + eav * pw[o];
        v = (v > 0.0f) ? v : 0.2f * v;         // leaky_relu 0.2
        acc += v * pa[o];
    }
    esc[i] = acc;
    atomicMaxF(&emax[(size_t)d * H + h], acc);
}

__global__ void k_emax_fix(float* emax, int n) {
    int i = blockIdx.x * blockDim.x + threadIdx.x;
    if (i >= n) return;
    float v = emax[i];
    if (!isfinite(v)) emax[i] = 0.0f;
}

__global__ void k_edge_exp(float* esc, const int* __restrict__ ei,
                           const float* __restrict__ emax, float* denom, int E, int H) {
    int i = blockIdx.x * blockDim.x + threadIdx.x;
    if (i >= E * H) return;
    int e = i / H, h = i - e * H;
    int d = ei[E + e];
    float ex = expf(esc[i] - emax[(size_t)d * H + h]);
    esc[i] = ex;
    atomicAdd(&denom[(size_t)d * H + h], ex);
}

__global__ void k_edge_agg(const float* __restrict__ esc, const float* __restrict__ denom,
                           const float* __restrict__ xl, const int* __restrict__ ei,
                           float* nxt, int E, int H, int O, int stride) {
    int e = blockIdx.x * blockDim.x + threadIdx.x;
    if (e >= E) return;
    int s = ei[e], d = ei[E + e];
    int HO = H * O;
    for (int h = 0; h < H; ++h) {
        float alpha = esc[(size_t)e * H + h] / (denom[(size_t)d * H + h] + 1e-16f);
        const float* pl = xl + (size_t)s * HO + h * O;
        float* pd = nxt + (size_t)d * stride + h * O;
        for (int o = 0; o < O; ++o) atomicAdd(&pd[o], pl[o] * alpha);
    }
}

// nxt: padded stride SPn (pads already zero); cur: padded stride SPc
__global__ void k_node_finish(float* nxt, const float* __restrict__ cur,
                              const float* __restrict__ cb,
                              int n, int HO, int ci, int SPn, int SPc, int doRelu) {
    int i = blockIdx.x * blockDim.x + threadIdx.x;
    if (i >= n) return;
    float* po = nxt + (size_t)i * SPn;
    for (int k = 0; k < HO; ++k) {
        float v = po[k] + cb[k];
        if (doRelu) v = fmaxf(v, 0.0f);
        po[k] = v;
    }
    const float* pc = cur + (size_t)i * SPc;
    for (int j = 0; j < ci; ++j) po[HO + j] = pc[j];
}

// ---------------------------------------------------------------- goal MLP
__global__ void k_dw_fc1(const float* __restrict__ goal, const float* __restrict__ W,
                         const float* __restrict__ b, float* hid, int G) {
    int i = blockIdx.x * blockDim.x + threadIdx.x;
    if (i >= G * 117) return;
    int g = i / 117, j = i - g * 117;
    float acc = b[j];
    for (int k = 0; k < 6; ++k) acc += goal[g * 6 + k] * W[k * 117 + j];
    hid[i] = fmaxf(acc, 0.0f);
}

__global__ void k_dw_fc2(const float* __restrict__ hid, const float* __restrict__ W,
                         const float* __restrict__ b, float* gw, int G) {
    int i = blockIdx.x * blockDim.x + threadIdx.x;
    if (i >= G * 117) return;
    int g = i / 117, j = i - g * 117;
    float acc = b[j];
    for (int k = 0; k < 117; ++k) acc += hid[g * 117 + k] * W[k * 117 + j];
    gw[i] = acc;
}

__global__ void k_scale(float* x, const float* __restrict__ gw,
                        const int* __restrict__ batch, int n, int F, int SP) {
    int i = blockIdx.x * blockDim.x + threadIdx.x;
    if (i >= n * F) return;
    int node = i / F, f = i - node * F;
    x[(size_t)node * SP + f] *= gw[batch[node] * F + f];
}

// ---------------------------------------------------------------- pooling
__global__ void k_pool(const float* __restrict__ x, const int* __restrict__ idxmap,
                       const int* __restrict__ seg, float* sum, float* cnt,
                       int n, int F, int SP) {
    int i = blockIdx.x * blockDim.x + threadIdx.x;
    if (i >= n) return;
    int row = idxmap ? idxmap[i] : i;
    int b = seg[i];
    atomicAdd(&cnt[b], 1.0f);
    const float* px = x + (size_t)row * SP;
    float* ps = sum + (size_t)b * F;
    for (int f = 0; f < F; ++f) atomicAdd(&ps[f], px[f]);
}

__global__ void k_emb(const float* __restrict__ sum, const float* __restrict__ cnt,
                      const float* __restrict__ W, const float* __restrict__ b,
                      float* emb, int G, int F, int Out) {
    int i = blockIdx.x * blockDim.x + threadIdx.x;
    if (i >= G * Out) return;
    int g = i / Out, o = i - g * Out;
    float c = fmaxf(cnt[g], 1.0f);
    float acc = b[o];
    for (int f = 0; f < F; ++f) {
        float s = sum[(size_t)g * F + f];
        acc += (s / c + s) * W[f * Out + o];    // mean-pool + add-pool
    }
    emb[i] = acc;
}

// ---------------------------------------------------------------- final head
__global__ void k_final(const float* __restrict__ x, const float* __restrict__ gemb,
                        const float* __restrict__ semb, const float* __restrict__ goal,
                        const int* __restrict__ batch, const int* __restrict__ amask,
                        const float* __restrict__ Wn, const float* __restrict__ bn,
                        const float* __restrict__ Ws, const float* __restrict__ bs,
                        const float* __restrict__ Wq, const float* __restrict__ bq,
                        const float* __restrict__ Wa, const float* __restrict__ ba,
                        const float* __restrict__ Wv, const float* __restrict__ bv,
                        float* out, int n, int SP) {
    int i = blockIdx.x * blockDim.x + threadIdx.x;
    if (i >= n) return;
    int b = batch[i];
    float conc[15];
    const float* px = x + (size_t)i * SP;
    for (int o = 0; o < 5; ++o) {
        float acc = bn[o];
        for (int f = 0; f < 127; ++f) acc += px[f] * Wn[f * 5 + o];
        conc[o] = acc;
    }
    for (int o = 0; o < 5; ++o) conc[5 + o]  = gemb[b * 5 + o];
    for (int o = 0; o < 5; ++o) conc[10 + o] = semb[b * 5 + o];
    float st[10];
    for (int o = 0; o < 10; ++o) {
        float acc = bs[o];
        for (int f = 0; f < 15; ++f) acc += conc[f] * Ws[f * 10 + o];
        st[o] = acc;
    }
    float qs[5];
    for (int o = 0; o < 5; ++o) {
        float acc = bq[o];
        for (int f = 0; f < 10; ++f) acc += st[f] * Wq[f * 5 + o];
        qs[o] = fmaxf(acc, 0.0f);
    }
    float adv = ba[0];
    for (int f = 0; f < 5; ++f) adv += qs[f] * Wa[f];
    for (int f = 0; f < 6; ++f) adv += goal[b * 6 + f] * Wa[5 + f];
    float qv = bv[0];
    for (int f = 0; f < 5; ++f) qv += qs[f] * Wv[f];
    qv += adv;
    out[i] = (amask[i] == 1) ? qv : -1e8f;
}

// ---------------------------------------------------------------- launcher
static inline int gs(int n) { return (n + 255) / 256; }

extern "C" void kernel_launch(void* const* d_in, const int* in_sizes, int n_in,
                              void* d_out, int out_size, void* d_ws, size_t ws_size,
                              hipStream_t stream) {
    (void)in_sizes; (void)n_in; (void)out_size; (void)ws_size;
    const int N = NNODES, E = NEDGES, G = NGRAPH, M = NSUB;

    // inputs in setup_inputs() insertion order; params dict expanded in
    // make_params insertion order (3 graphnorm + 5*7 conv + 18 linear leaves).
    const float* x0        = (const float*)d_in[0];
    const float* edge_attr = (const float*)d_in[1];
    const float* goal      = (const float*)d_in[2];
    const float* gn_w  = (const float*)d_in[3];
    const float* gn_b  = (const float*)d_in[4];
    const float* gn_ms = (const float*)d_in[5];
    const float *Wl[5], *bl[5], *Wr[5], *br[5], *We[5], *att[5], *cb[5];
    for (int i = 0; i < 5; ++i) {
        int base = 6 + 7 * i;
        Wl[i]  = (const float*)d_in[base + 0];
        bl[i]  = (const float*)d_in[base + 1];
        Wr[i]  = (const float*)d_in[base + 2];
        br[i]  = (const float*)d_in[base + 3];
        We[i]  = (const float*)d_in[base + 4];
        att[i] = (const float*)d_in[base + 5];
        cb[i]  = (const float*)d_in[base + 6];
    }
    const float* fc1W = (const float*)d_in[41]; const float* fc1b = (const float*)d_in[42];
    const float* fc2W = (const float*)d_in[43]; const float* fc2b = (const float*)d_in[44];
    const float* nW  = (const float*)d_in[45]; const float* nB  = (const float*)d_in[46];
    const float* gW  = (const float*)d_in[47]; const float* gB  = (const float*)d_in[48];
    const float* sW  = (const float*)d_in[49]; const float* sB  = (const float*)d_in[50];
    const float* stW = (const float*)d_in[51]; const float* stB = (const float*)d_in[52];
    const float* qsW = (const float*)d_in[53]; const float* qsB = (const float*)d_in[54];
    const float* gaW = (const float*)d_in[55]; const float* gaB = (const float*)d_in[56];
    const float* qvW = (const float*)d_in[57]; const float* qvB = (const float*)d_in[58];
    const int* ei     = (const int*)d_in[59];
    const int* batch  = (const int*)d_in[60];
    const int* amask  = (const int*)d_in[61];
    const int* vsub   = (const int*)d_in[62];
    const int* sbatch = (const int*)d_in[63];

    float* ws = (float*)d_ws;
    size_t o = 0;
    float* xA   = ws + o; o += (size_t)N * 128;
    float* xB   = ws + o; o += (size_t)N * 128;
    float* xl   = ws + o; o += (size_t)N * 48;
    float* xr   = ws + o; o += (size_t)N * 48;
    float* esc  = ws + o; o += (size_t)E * 2;
    float* emax = ws + o; o += (size_t)N * 2;
    float* den  = ws + o; o += (size_t)N * 2;
    float* gsum = ws + o; o += (size_t)G * 128;
    float* gcnt = ws + o; o += G;
    float* gvar = ws + o; o += (size_t)G * 8;
    float* ssum = ws + o; o += (size_t)G * 128;
    float* scnt = ws + o; o += G;
    float* hid  = ws + o; o += (size_t)G * 128;
    float* gwv  = ws + o; o += (size_t)G * 128;
    float* gemb = ws + o; o += (size_t)G * 8;
    float* semb = ws + o; o += (size_t)G * 8;

    // padded feature strides per layer (multiple of 4 for WMMA K-steps)
    const int CI[5]   = {7, 27, 67, 97, 117};
    const int CIP[5]  = {8, 28, 68, 100, 120};
    const int CO[5]   = {10, 20, 30, 20, 10};
    const int HH[5]   = {2, 2, 1, 1, 1};
    const int NCIP[5] = {28, 68, 100, 120, 128};

    // ---- graph_norm (F=7, output padded stride 8, pad col zeroed)
    k_fill<<<gs(G * 7), 256, 0, stream>>>(gsum, 0.f, G * 7);
    k_fill<<<gs(G), 256, 0, stream>>>(gcnt, 0.f, G);
    k_fill<<<gs(G * 7), 256, 0, stream>>>(gvar, 0.f, G * 7);
    k_fill<<<gs(N * 8), 256, 0, stream>>>(xA, 0.f, N * 8);
    k_gn_sum<<<gs(N), 256, 0, stream>>>(x0, batch, gsum, gcnt, N, 7);
    k_gn_mean<<<gs(G * 7), 256, 0, stream>>>(gsum, gcnt, G, 7);
    k_gn_var<<<gs(N), 256, 0, stream>>>(x0, batch, gsum, gn_ms, xA, gvar, N, 7, 8);
    k_gn_fin<<<gs(N), 256, 0, stream>>>(xA, batch, gvar, gcnt, gn_w, gn_b, N, 7, 8);

    // ---- GATv2 layers
    float* cur = xA;
    float* nxt = xB;
    for (int i = 0; i < 5; ++i) {
        int ci = CI[i], cip = CIP[i], Hh = HH[i], Oc = CO[i];
        int HO = Hh * Oc, ncip = NCIP[i];
        if (i == 4) {  // dynamic goal weighting before conv5
            k_dw_fc1<<<gs(G * 117), 256, 0, stream>>>(goal, fc1W, fc1b, hid, G);
            k_dw_fc2<<<gs(G * 117), 256, 0, stream>>>(hid, fc2W, fc2b, gwv, G);
            k_scale<<<gs(N * 117), 256, 0, stream>>>(cur, gwv, batch, N, 117, cip);
        }
        dim3 grid((N / 16 + 7) / 8, (HO + 15) / 16);
        k_gemm_wmma<<<grid, 256, 0, stream>>>(cur, Wl[i], bl[i], xl, N, ci, cip, HO);
        k_gemm_wmma<<<grid, 256, 0, stream>>>(cur, Wr[i], br[i], xr, N, ci, cip, HO);

        k_fill<<<gs(N * Hh), 256, 0, stream>>>(emax, -INFINITY, N * Hh);
        k_fill<<<gs(N * Hh), 256, 0, stream>>>(den, 0.f, N * Hh);
        k_edge_score<<<gs(E * Hh), 256, 0, stream>>>(xl, xr, edge_attr, ei, We[i], att[i],
                                                     esc, emax, E, Hh, Oc);
        k_emax_fix<<<gs(N * Hh), 256, 0, stream>>>(emax, N * Hh);
        k_edge_exp<<<gs(E * Hh), 256, 0, stream>>>(esc, ei, emax, den, E, Hh);
        k_fill<<<gs(N * ncip), 256, 0, stream>>>(nxt, 0.f, N * ncip);
        k_edge_agg<<<gs(E), 256, 0, stream>>>(esc, den, xl, ei, nxt, E, Hh, Oc, ncip);
        k_node_finish<<<gs(N), 256, 0, stream>>>(nxt, cur, cb[i], N, HO, ci, ncip, cip,
                                                 i < 4 ? 1 : 0);
        float* t = cur; cur = nxt; nxt = t;
    }

    // ---- pooling + embeddings (x: [N,127] stride 128)
    k_fill<<<gs(G * 127), 256, 0, stream>>>(gsum, 0.f, G * 127);
    k_fill<<<gs(G), 256, 0, stream>>>(gcnt, 0.f, G);
    k_fill<<<gs(G * 127), 256, 0, stream>>>(ssum, 0.f, G * 127);
    k_fill<<<gs(G), 256, 0, stream>>>(scnt, 0.f, G);
    k_pool<<<gs(N), 256, 0, stream>>>(cur, (const int*)nullptr, batch, gsum, gcnt, N, 127, 128);
    k_pool<<<gs(M), 256, 0, stream>>>(cur, vsub, sbatch, ssum, scnt, M, 127, 128);
    k_emb<<<gs(G * 5), 256, 0, stream>>>(gsum, gcnt, gW, gB, gemb, G, 127, 5);
    k_emb<<<gs(G * 5), 256, 0, stream>>>(ssum, scnt, sW, sB, semb, G, 127, 5);

    // ---- final per-node head
    k_final<<<gs(N), 256, 0, stream>>>(cur, gemb, semb, goal, batch, amask,
                                       nW, nB, stW, stB, qsW, qsB, gaW, gaB, qvW, qvB,
                                       (float*)d_out, N, 128);
}